// EmbeddingMul_73916387164601
// MI455X (gfx1250) — compile-verified
//
#include <hip/hip_runtime.h>
#include <hip/hip_bf16.h>

// Embedding gather: out[row, :] = weight[input[row], :]
//   rows = BPTT*BSIZE = 16384, EMB = 512 f32 -> 2048 bytes per row.
// Pure data movement (~64 MB total) -> HBM-bound, ~2.7 us floor @ 23.3 TB/s.
// Implemented with CDNA5 async global<->LDS DMA (ASYNCcnt-tracked), one wave
// per row, 4 x 512B b128-wide chunks per row, fully coalesced.

#define EMB_FLOATS 512
#define ROW_BYTES  (EMB_FLOATS * 4)     // 2048 B per row
#define WAVES_PER_BLOCK 8
#define THREADS_PER_BLOCK (WAVES_PER_BLOCK * 32)

__global__ __launch_bounds__(THREADS_PER_BLOCK)
void EmbeddingMul_gather_async(const int* __restrict__ input,
                               const float* __restrict__ weight,
                               float* __restrict__ out,
                               int nrows)
{
    __shared__ __align__(16) char smem[WAVES_PER_BLOCK * ROW_BYTES];

    const int wave = (int)(threadIdx.x >> 5);
    const int lane = (int)(threadIdx.x & 31);
    const int row  = (int)blockIdx.x * WAVES_PER_BLOCK + wave;  // wave-uniform
    if (row >= nrows) return;  // whole wave exits together (row is uniform)

    // Wave-uniform embedding index.
    const int idx = __builtin_amdgcn_readfirstlane(input[row]);

    // Per-lane addresses. The instruction offset (0/512/1024/1536) is applied
    // by HW to BOTH the LDS address and the global address, so one base pair
    // covers the whole 2048-B row in 4 async b128 ops per lane.
    const unsigned lds_off =
        (unsigned)(uintptr_t)(&smem[wave * ROW_BYTES + lane * 16]);
    const unsigned long long src =
        (unsigned long long)(uintptr_t)weight
        + (unsigned long long)(unsigned)idx * (unsigned long long)ROW_BYTES
        + (unsigned long long)(lane * 16);
    const unsigned long long dst =
        (unsigned long long)(uintptr_t)out
        + (unsigned long long)(unsigned)row * (unsigned long long)ROW_BYTES
        + (unsigned long long)(lane * 16);

    // Global -> LDS: 4 x 512B async DMA chunks (ASYNCcnt += 4).
    asm volatile(
        "global_load_async_to_lds_b128 %0, %1, off\n\t"
        "global_load_async_to_lds_b128 %0, %1, off offset:512\n\t"
        "global_load_async_to_lds_b128 %0, %1, off offset:1024\n\t"
        "global_load_async_to_lds_b128 %0, %1, off offset:1536"
        :
        : "v"(lds_off), "v"(src)
        : "memory");

    // Loads complete (LDS written) before the stores read it.
    asm volatile("s_wait_asynccnt 0" ::: "memory");

    // LDS -> Global out: 4 x 512B async DMA chunks.
    asm volatile(
        "global_store_async_from_lds_b128 %1, %0, off\n\t"
        "global_store_async_from_lds_b128 %1, %0, off offset:512\n\t"
        "global_store_async_from_lds_b128 %1, %0, off offset:1024\n\t"
        "global_store_async_from_lds_b128 %1, %0, off offset:1536"
        :
        : "v"(lds_off), "v"(dst)
        : "memory");

    // Drain before wave teardown (S_ENDPGM also implicitly waits idle).
    asm volatile("s_wait_asynccnt 0" ::: "memory");
}

extern "C" void kernel_launch(void* const* d_in, const int* in_sizes, int n_in,
                              void* d_out, int out_size, void* d_ws, size_t ws_size,
                              hipStream_t stream)
{
    (void)n_in; (void)d_ws; (void)ws_size; (void)out_size;

    const int*   input  = (const int*)d_in[0];    // (BPTT, BSIZE) int32, flat
    const float* weight = (const float*)d_in[1];  // (VOCAB, 512) f32
    float*       out    = (float*)d_out;          // (BPTT, BSIZE, 512) f32

    const int nrows  = in_sizes[0];               // 16384
    const int blocks = (nrows + WAVES_PER_BLOCK - 1) / WAVES_PER_BLOCK;

    EmbeddingMul_gather_async<<<blocks, THREADS_PER_BLOCK, 0, stream>>>(
        input, weight, out, nrows);
}